// FuncMod_25340307047168
// MI455X (gfx1250) — compile-verified
//
#include <hip/hip_runtime.h>
#include <math.h>

// ---------------- problem constants (match reference) ----------------
#define B_N    16384
#define IN_C   128
#define CHN    512
#define EMBED  6240
#define NUMEMB 512
#define DEC_IN 64
#define DEC_H  64
#define DATA_Y 32
#define KC     32          // K-chunk staged in LDS (all K here are multiples of 32)

typedef __attribute__((ext_vector_type(2))) float v2f;
typedef __attribute__((ext_vector_type(8))) float v8f;

// =====================================================================
// fp32 WMMA GEMM:  C[M,N] = act(A[M,K] @ B[K,N] + bias)
// block = 256 threads (8 waves); block tile = 128 rows x 64 cols.
// Wave w owns rows [by*128 + w*16, +16) and all 64 cols (4 WMMA tiles
// sharing one A fragment).  B panel (KC x 64) is cooperatively staged
// in LDS once per block (8x dedup vs per-wave global loads), stored
// K-PAIR INTERLEAVED:  Bs[(k/2)*128 + 2*c + (k&1)]
// so a lane's WMMA B fragment (B[k][c], B[k+1][c]) is one contiguous
// 8B-aligned pair -> single ds_load_b64 into an even VGPR pair, no movs.
// Fragment layouts per CDNA5 ISA 7.12.2 (V_WMMA_F32_16X16X4_F32):
//   A 16x4 : lane<16 holds A[M=lane][k0],A[M][k0+1]; lanes 16-31 hold k0+2,k0+3
//   B 4x16 : vgpr0 = row k0 (lanes 0-15) / row k0+2 (lanes 16-31); vgpr1 = +1
//   D 16x16: vgpr r = row r (lanes 0-15) / row r+8 (lanes 16-31), col = lane%16
// =====================================================================
__global__ __launch_bounds__(256)
void wmma_gemm_bias(const float* __restrict__ A, const float* __restrict__ Bm,
                    const float* __restrict__ bias, float* __restrict__ C,
                    int N, int K, int relu)
{
    __shared__ float Bs[(KC / 2) * 128];    // 8 KB B panel, pair-interleaved
    const int lane = threadIdx.x & 31;
    const int wave = threadIdx.x >> 5;
    const int half = lane >> 4;             // 0: K pair {0,1}, 1: K pair {2,3}
    const int l16  = lane & 15;
    const int row0 = blockIdx.y * 128 + wave * 16;
    const int col0 = blockIdx.x * 64;

    v8f acc[4] = {{}, {}, {}, {}};

    const float* Arow = A + (size_t)(row0 + l16) * K + half * 2;

    for (int kc = 0; kc < K; kc += KC) {
        // ---- stage B[kc..kc+KC) x [col0..col0+64) into LDS (pair layout) ----
        for (int e = threadIdx.x; e < KC * 64; e += 256) {
            const int kk = e >> 6;          // 0..KC-1
            const int cc = e & 63;
            int c = col0 + cc;
            if (c > N - 1) c = N - 1;       // clamp for the N%64 tail block
            Bs[(kk >> 1) * 128 + cc * 2 + (kk & 1)] =
                Bm[(size_t)(kc + kk) * N + c];
        }
        __syncthreads();

        if (kc + KC < K)                     // speculative prefetch of next A panel
            __builtin_prefetch(Arow + kc + KC, 0, 1);

        for (int k0 = 0; k0 < KC; k0 += 4) {
            v2f a = *(const v2f*)(Arow + kc + k0);             // global b64
            // k0 + half*2 is even -> pair row = k0/2 + half
            const float* bs = &Bs[(k0 / 2 + half) * 128];
#pragma unroll
            for (int t = 0; t < 4; ++t) {
                v2f b = *(const v2f*)(&bs[(t * 16 + l16) * 2]); // ds b64
                acc[t] = __builtin_amdgcn_wmma_f32_16x16x4_f32(
                    false, a, false, b, (short)0, acc[t], false, false);
            }
        }
        __syncthreads();
    }

    // ---- epilogue: bias + optional ReLU, per-tile store guard (uniform) ----
#pragma unroll
    for (int t = 0; t < 4; ++t) {
        if (col0 + t * 16 < N) {             // N%16==0 -> tiles fully valid or not
            const int c = col0 + t * 16 + l16;
            const float bi = bias ? bias[c] : 0.0f;
#pragma unroll
            for (int r = 0; r < 8; ++r) {
                const int row = row0 + half * 8 + r;
                float v = acc[t][r] + bi;
                if (relu) v = fmaxf(v, 0.0f);
                C[(size_t)row * N + c] = v;
            }
        }
    }
}

// ------------- order-preserving float -> uint mapping ----------------
__device__ __forceinline__ unsigned int ordkey(float f)
{
    unsigned int u = __float_as_uint(f);
    return (u & 0x80000000u) ? ~u : (u | 0x80000000u);
}

// =====================================================================
// Fused VQ scoring + argmin + index emit + histogram.
// score[b,n] = ||e_n||^2 - 2*(pre_f[b].e_n)  (||pre_f||^2 is row-constant).
// One block per 16-row strip of pre_f, covering ALL 512 codewords:
//   wave w -> cols [w*64, +64) as 4 WMMA tiles sharing one A fragment.
// The A strip (16 x KC) is staged in LDS once per chunk -> pre_f streams
// from HBM exactly once (409 MB total).  embed (12.8 MB) stays L2-resident.
// Argmin: per-lane pack (ordered_float<<32 | col) -> min over 4 tiles ->
// shfl_xor min across the 16 lanes of each half -> LDS min across 8 waves.
// Min of packed keys == first minimal column, matching jnp.argmin ties.
// =====================================================================
__global__ __launch_bounds__(256)
void wmma_score_argmin(const float* __restrict__ A,      // pre_f [B, EMBED]
                       const float* __restrict__ Em,     // embed [EMBED, NUMEMB]
                       const float* __restrict__ norms,  // [NUMEMB]
                       int* __restrict__ ind,
                       float* __restrict__ out_ind,
                       unsigned int* __restrict__ counts)
{
    __shared__ float As[16 * 34];                     // stride 34: pad -> b64 reads
    __shared__ unsigned long long red[8][16];
    const int lane = threadIdx.x & 31;
    const int wave = threadIdx.x >> 5;
    const int half = lane >> 4;
    const int l16  = lane & 15;
    const int row0  = blockIdx.x * 16;
    const int colw0 = wave * 64;

    v8f acc[4] = {{}, {}, {}, {}};

    for (int kc = 0; kc < EMBED; kc += KC) {
        // ---- stage A[row0..+16) x [kc..kc+KC) into LDS (coalesced) ----
        for (int e = threadIdx.x; e < 16 * KC; e += 256) {
            const int r  = e >> 5;        // 0..15
            const int kk = e & 31;        // 0..31
            As[r * 34 + kk] = A[(size_t)(row0 + r) * EMBED + kc + kk];
        }
        __syncthreads();

        for (int k0 = 0; k0 < KC; k0 += 4) {
            v2f a = *(const v2f*)(&As[l16 * 34 + k0 + half * 2]);   // ds b64
            const float* Bk = Em + (size_t)(kc + k0 + half * 2) * NUMEMB
                                 + colw0 + l16;
#pragma unroll
            for (int t = 0; t < 4; ++t) {
                v2f b;
                b.x = Bk[t * 16];
                b.y = Bk[NUMEMB + t * 16];
                acc[t] = __builtin_amdgcn_wmma_f32_16x16x4_f32(
                    false, a, false, b, (short)0, acc[t], false, false);
            }
        }
        __syncthreads();
    }

    // ---- per-row argmin across this wave's 64 columns ----
    float nrm[4];
#pragma unroll
    for (int t = 0; t < 4; ++t) nrm[t] = norms[colw0 + t * 16 + l16];

#pragma unroll
    for (int r = 0; r < 8; ++r) {
        unsigned long long key = ~0ULL;
#pragma unroll
        for (int t = 0; t < 4; ++t) {
            const float v = nrm[t] - 2.0f * acc[t][r];
            const unsigned long long k2 =
                ((unsigned long long)ordkey(v) << 32)
                | (unsigned int)(colw0 + t * 16 + l16);
            key = (k2 < key) ? k2 : key;
        }
        // min across the 16 lanes of this half (masks < 16 keep halves intact)
        for (int m = 1; m < 16; m <<= 1) {
            const unsigned long long o = __shfl_xor(key, m, 32);
            key = (o < key) ? o : key;
        }
        if (l16 == 0) red[wave][half * 8 + r] = key;
    }
    __syncthreads();

    // ---- combine across the 8 waves, emit index + histogram ----
    if (threadIdx.x < 16) {
        unsigned long long key = red[0][threadIdx.x];
#pragma unroll
        for (int w = 1; w < 8; ++w) {
            const unsigned long long o = red[w][threadIdx.x];
            key = (o < key) ? o : key;
        }
        const int k = (int)(unsigned int)(key & 0xFFFFFFFFull);
        ind[row0 + threadIdx.x]     = k;
        out_ind[row0 + threadIdx.x] = (float)k;
        atomicAdd(&counts[k], 1u);
    }
}

// ---------------- small support kernels ------------------------------
__global__ void init_ws_kernel(unsigned int* counts, double* accum)
{
    const int i = blockIdx.x * blockDim.x + threadIdx.x;
    if (i < NUMEMB) counts[i] = 0u;
    if (i == 0)     *accum = 0.0;
}

__global__ void embed_norms_kernel(const float* __restrict__ E,
                                   float* __restrict__ norms)
{
    const int n = threadIdx.x;  // 512 threads
    float s = 0.0f;
    for (int k = 0; k < EMBED; ++k) {
        const float e = E[(size_t)k * NUMEMB + n];
        s = fmaf(e, e, s);
    }
    norms[n] = s;
}

__global__ __launch_bounds__(256)
void diff_kernel(const float* __restrict__ pre_f, const float* __restrict__ E,
                 const int* __restrict__ ind, double* __restrict__ accum)
{
    __shared__ float red[256];
    const int b = blockIdx.x;
    const int k = ind[b];
    const float* row = pre_f + (size_t)b * EMBED;
    float s = 0.0f;
    for (int j = threadIdx.x; j < EMBED; j += 256) {
        const float d = E[(size_t)j * NUMEMB + k] - row[j];
        s = fmaf(d, d, s);
    }
    red[threadIdx.x] = s;
    __syncthreads();
    for (int off = 128; off > 0; off >>= 1) {
        if (threadIdx.x < off) red[threadIdx.x] += red[threadIdx.x + off];
        __syncthreads();
    }
    if (threadIdx.x == 0) atomicAdd(accum, (double)red[0]);
}

__global__ void finalize_scalars_kernel(const unsigned int* __restrict__ counts,
                                        const double* __restrict__ accum,
                                        float* __restrict__ out_diff,
                                        float* __restrict__ out_perp)
{
    __shared__ float red[NUMEMB];
    const int n = threadIdx.x;  // 512 threads
    const float p = (float)counts[n] * (1.0f / (float)B_N);
    red[n] = p * logf(p + 1e-10f);
    __syncthreads();
    for (int off = NUMEMB / 2; off > 0; off >>= 1) {
        if (n < off) red[n] += red[n + off];
        __syncthreads();
    }
    if (n == 0) {
        *out_perp = expf(-red[0]);
        *out_diff = (float)(*accum / ((double)B_N * (double)EMBED));
    }
}

// ---------------- per-sample expert MLP decoder ----------------------
__global__ __launch_bounds__(64)
void decoder_kernel(const float* __restrict__ ex, const int* __restrict__ ind,
                    const float* __restrict__ W1, const float* __restrict__ b1,
                    const float* __restrict__ W2, const float* __restrict__ b2,
                    float* __restrict__ out)
{
    __shared__ float xs[DEC_IN];
    __shared__ float hs[DEC_H];
    const int b = blockIdx.x;
    const int t = threadIdx.x;
    const int k = ind[b];
    xs[t] = ex[(size_t)b * DEC_IN + t];
    __syncthreads();
    const float* w1 = W1 + (size_t)k * DEC_IN * DEC_H;
    float h = b1[(size_t)k * DEC_H + t];
#pragma unroll 8
    for (int i = 0; i < DEC_IN; ++i)
        h = fmaf(xs[i], w1[(size_t)i * DEC_H + t], h);
    hs[t] = fmaxf(h, 0.0f);
    __syncthreads();
    if (t < DATA_Y) {
        const float* w2 = W2 + (size_t)k * DEC_H * DATA_Y;
        float o = b2[(size_t)k * DATA_Y + t];
#pragma unroll 8
        for (int i = 0; i < DEC_H; ++i)
            o = fmaf(hs[i], w2[(size_t)i * DATA_Y + t], o);
        out[(size_t)b * DATA_Y + t] = o;
    }
}

// =====================================================================
extern "C" void kernel_launch(void* const* d_in, const int* in_sizes, int n_in,
                              void* d_out, int out_size, void* d_ws, size_t ws_size,
                              hipStream_t stream)
{
    const float* x   = (const float*)d_in[0];
    const float* Wf1 = (const float*)d_in[1];
    const float* bf1 = (const float*)d_in[2];
    const float* Wf2 = (const float*)d_in[3];
    const float* bf2 = (const float*)d_in[4];
    const float* Wx1 = (const float*)d_in[5];
    const float* bx1 = (const float*)d_in[6];
    const float* Wx2 = (const float*)d_in[7];
    const float* bx2 = (const float*)d_in[8];
    const float* Em  = (const float*)d_in[9];
    const float* W1  = (const float*)d_in[10];
    const float* b1  = (const float*)d_in[11];
    const float* W2  = (const float*)d_in[12];
    const float* b2  = (const float*)d_in[13];

    // output layout: dec[B,32] | diff | embed_ind[B] (as float) | perplexity
    float* out      = (float*)d_out;
    float* out_dec  = out;
    float* out_diff = out + (size_t)B_N * DATA_Y;
    float* out_ind  = out_diff + 1;
    float* out_perp = out_ind + B_N;

    // workspace carve-out (256B aligned slices)
    char*  ws  = (char*)d_ws;
    size_t off = 0;
    auto alloc = [&](size_t bytes) -> void* {
        void* p = ws + off;
        off += (bytes + 255) & ~(size_t)255;
        return p;
    };
    float* h_f   = (float*)alloc((size_t)B_N * CHN * 4);     //  32 MB
    float* h_x   = (float*)alloc((size_t)B_N * CHN * 4);     //  32 MB
    float* pre_f = (float*)alloc((size_t)B_N * EMBED * 4);   // 409 MB
    float* ex    = (float*)alloc((size_t)B_N * DEC_IN * 4);  //   4 MB
    float* norms = (float*)alloc((size_t)NUMEMB * 4);
    int*          ind    = (int*)alloc((size_t)B_N * 4);
    unsigned int* counts = (unsigned int*)alloc((size_t)NUMEMB * 4);
    double*       accum  = (double*)alloc(8);
    (void)ws_size; (void)in_sizes; (void)n_in; (void)out_size;

    const dim3 blk(256);

    init_ws_kernel<<<2, 256, 0, stream>>>(counts, accum);
    embed_norms_kernel<<<1, NUMEMB, 0, stream>>>(Em, norms);

    // enc_f: x@Wf1+bf1, relu -> h_f ; h_f@Wf2+bf2 -> pre_f
    wmma_gemm_bias<<<dim3((CHN + 63) / 64, B_N / 128), blk, 0, stream>>>(
        x, Wf1, bf1, h_f, CHN, IN_C, 1);
    wmma_gemm_bias<<<dim3((EMBED + 63) / 64, B_N / 128), blk, 0, stream>>>(
        h_f, Wf2, bf2, pre_f, EMBED, CHN, 0);

    // enc_x: x@Wx1+bx1, relu -> h_x ; h_x@Wx2+bx2 -> ex
    wmma_gemm_bias<<<dim3((CHN + 63) / 64, B_N / 128), blk, 0, stream>>>(
        x, Wx1, bx1, h_x, CHN, IN_C, 1);
    wmma_gemm_bias<<<dim3((DEC_IN + 63) / 64, B_N / 128), blk, 0, stream>>>(
        h_x, Wx2, bx2, ex, DEC_IN, CHN, 0);

    // VQ: fused score GEMM + argmin + index emit + histogram (single pass
    // over pre_f: one 16-row strip per block covering all 512 codewords)
    wmma_score_argmin<<<B_N / 16, blk, 0, stream>>>(
        pre_f, Em, norms, ind, out_ind, counts);

    diff_kernel<<<B_N, 256, 0, stream>>>(pre_f, Em, ind, accum);
    finalize_scalars_kernel<<<1, NUMEMB, 0, stream>>>(
        counts, accum, out_diff, out_perp);
    decoder_kernel<<<B_N, 64, 0, stream>>>(ex, ind, W1, b1, W2, b2, out_dec);
}